// GraphSAGE_81226421502183
// MI455X (gfx1250) — compile-verified
//
#include <hip/hip_runtime.h>

typedef __attribute__((ext_vector_type(2))) float v2f;
typedef __attribute__((ext_vector_type(8))) float v8f;

#define IN_DIM 165
#define HID    64

// ---------------- utility kernels ----------------

__global__ void zero_f32(float* __restrict__ p, int n) {
  int i = blockIdx.x * blockDim.x + threadIdx.x;
  if (i < n) p[i] = 0.0f;
}

__global__ void degree_kernel(const int* __restrict__ dst, float* __restrict__ cnt, int E) {
  int e = blockIdx.x * blockDim.x + threadIdx.x;
  if (e < E) atomicAdd(&cnt[dst[e]], 1.0f);
}

// Pack W[K,64] -> Wp[KPAD/2][64][2]: K-pairs interleaved per column, zero padded.
// Lets the GEMM fetch a B fragment (two K-values, one column) as one b64 load.
__global__ void pack_weights(const float* __restrict__ W, float* __restrict__ Wp,
                             int K, int KPAD) {
  int i = blockIdx.x * blockDim.x + threadIdx.x;   // over (KPAD/2)*64
  int total = (KPAD >> 1) * HID;
  if (i >= total) return;
  int kp = i >> 6;          // K-pair index
  int c  = i & 63;          // column
  int k  = kp << 1;
  float v0 = (k     < K) ? W[(size_t)k * HID + c]       : 0.0f;
  float v1 = (k + 1 < K) ? W[(size_t)(k + 1) * HID + c] : 0.0f;
  Wp[((size_t)i << 1) + 0] = v0;
  Wp[((size_t)i << 1) + 1] = v1;
}

// ---------------- dual WMMA GEMM, 32-row M tile ----------------
// OUTL[N,64] = X[N,K] @ WL[K,64] ; OUTR[N,64] = X[N,K] @ WR[K,64]
// Block = 256 threads = 8 waves. Waves 0..3 -> WL cols [16w,16w+16),
// waves 4..7 -> WR cols. Each wave keeps two 16x16 accumulators covering the
// block's 32 X-rows staged in LDS (zero padded to KPAD, mult of 4).
// K/KPAD are compile-time so the fill addressing and K loop fully constant-fold;
// inner loop: 2x ds_load_b64 (A) + 1x global_load_b64 (B) + 2x WMMA, no branches.
template <int K, int KPAD, bool CHECK>
__global__ __launch_bounds__(256)
void gemm_dual_wmma(const float* __restrict__ X,
                    const float* __restrict__ WLp, const float* __restrict__ WRp,
                    float* __restrict__ OUTL, float* __restrict__ OUTR, int N) {
  __shared__ float Xs[32 * KPAD];
  const int tid = threadIdx.x;
  const int n0  = blockIdx.x * 32;

  // fill LDS tile (zero pad columns K..KPAD; rows beyond N if CHECK)
  for (int i = tid; i < 32 * KPAD; i += 256) {
    int r = i / KPAD;
    int c = i - r * KPAD;
    int gr = n0 + r;
    bool ok = (c < K) && (!CHECK || gr < N);
    Xs[i] = ok ? X[(size_t)gr * K + c] : 0.0f;
  }
  __syncthreads();

  const int wave = tid >> 5;   // 0..7
  const int lane = tid & 31;
  const int half = lane >> 4;  // 0: K {k,k+1}/rows M 0-7 ; 1: K {k+2,k+3}/rows M 8-15
  const int l16  = lane & 15;
  const float* __restrict__ Wp  = (wave < 4) ? WLp : WRp;
  float* __restrict__       OUT = (wave < 4) ? OUTL : OUTR;
  const int col = ((wave & 3) << 4) + l16;

  const float* __restrict__ A0 = Xs + l16 * KPAD;        // rows n0..n0+15
  const float* __restrict__ A1 = Xs + (16 + l16) * KPAD; // rows n0+16..n0+31
  const float* __restrict__ Bp = Wp + ((size_t)col << 1);

  v8f acc0 = {};
  v8f acc1 = {};
#pragma unroll 2
  for (int k = 0; k < KPAD; k += 4) {
    const int ka = k + 2 * half;  // even
    v2f a0 = *(const v2f*)(A0 + ka);
    v2f a1 = *(const v2f*)(A1 + ka);
    v2f b  = *(const v2f*)(Bp + ((size_t)(ka >> 1) << 7)); // pair (ka/2), stride 128 floats
    acc0 = __builtin_amdgcn_wmma_f32_16x16x4_f32(false, a0, false, b,
                                                 (short)0, acc0, false, false);
    acc1 = __builtin_amdgcn_wmma_f32_16x16x4_f32(false, a1, false, b,
                                                 (short)0, acc1, false, false);
  }

  // D layout: VGPR r -> M = r + 8*half, N = col
#pragma unroll
  for (int r = 0; r < 8; ++r) {
    int row0 = n0 + r + 8 * half;
    int row1 = row0 + 16;
    if (!CHECK || row0 < N) OUT[(size_t)row0 * HID + col] = acc0[r];
    if (!CHECK || row1 < N) OUT[(size_t)row1 * HID + col] = acc1[r];
  }
}

// ---------------- edge scatter (projected features, 64-dim) ----------------
__global__ void scatter_feat(const float* __restrict__ P, const int* __restrict__ src,
                             const int* __restrict__ dst, float* __restrict__ agg, int E) {
  long long gid = (long long)blockIdx.x * blockDim.x + threadIdx.x;
  int e = (int)(gid >> 4);
  if (e >= E) return;
  int q = ((int)gid & 15) << 2;
  int s = src[e];
  int t = dst[e];
  const float4 v = *(const float4*)(P + (size_t)s * HID + q);
  float* b = agg + (size_t)t * HID + q;
  atomicAdd(b + 0, v.x);
  atomicAdd(b + 1, v.y);
  atomicAdd(b + 2, v.z);
  atomicAdd(b + 3, v.w);
}

// ---------------- mean + bias + root path + ReLU ----------------
__global__ void combine_relu(const float* __restrict__ agg, const float* __restrict__ cnt,
                             const float* __restrict__ bias, const float* __restrict__ R,
                             float* __restrict__ H, int N) {
  int i = blockIdx.x * blockDim.x + threadIdx.x;
  if (i >= N * HID) return;
  int n = i >> 6;
  int d = i & 63;
  float c = fmaxf(cnt[n], 1.0f);
  float v = agg[i] / c + bias[d] + R[i];
  H[i] = fmaxf(v, 0.0f);
}

// ---------------- classifier 64 -> 2 ----------------
__global__ void classifier(const float* __restrict__ H, const float* __restrict__ Wc,
                           const float* __restrict__ bc, float* __restrict__ out, int N) {
  int n = blockIdx.x * blockDim.x + threadIdx.x;
  if (n >= N) return;
  float a0 = bc[0], a1 = bc[1];
  const float* h = H + (size_t)n * HID;
#pragma unroll
  for (int d = 0; d < HID; ++d) {
    float hv = h[d];
    a0 += hv * Wc[d * 2 + 0];
    a1 += hv * Wc[d * 2 + 1];
  }
  out[n * 2 + 0] = a0;
  out[n * 2 + 1] = a1;
}

// ---------------- host orchestration ----------------

extern "C" void kernel_launch(void* const* d_in, const int* in_sizes, int n_in,
                              void* d_out, int out_size, void* d_ws, size_t ws_size,
                              hipStream_t stream) {
  const float* x   = (const float*)d_in[0];
  const int*   ei  = (const int*)d_in[1];
  const float* W1l = (const float*)d_in[2];
  const float* b1  = (const float*)d_in[3];
  const float* W1r = (const float*)d_in[4];
  const float* W2l = (const float*)d_in[5];
  const float* b2  = (const float*)d_in[6];
  const float* W2r = (const float*)d_in[7];
  const float* Wc  = (const float*)d_in[8];
  const float* bc  = (const float*)d_in[9];

  const int N = in_sizes[0] / IN_DIM;   // 100000
  const int E = in_sizes[1] / 2;        // 1600000
  const int* src = ei;
  const int* dst = ei + E;

  const int NH = N * HID;
  constexpr int K1 = IN_DIM, K1P = 168; // pad 165 -> 168 (mult of 4)
  constexpr int K2 = HID,    K2P = HID;

  // workspace layout (floats): A | B | C | H | CNT | packed weights
  float* A    = (float*)d_ws;  // projected-neighbor features (P)
  float* B    = A + NH;        // root-path features (R)
  float* C    = B + NH;        // aggregation accumulator
  float* H    = C + NH;        // hidden activations
  float* CNT  = H + NH;        // in-degree
  float* Wp1l = CNT + N;       // 168*64
  float* Wp1r = Wp1l + K1P * HID;
  float* Wp2l = Wp1r + K1P * HID;  // 64*64
  float* Wp2r = Wp2l + K2P * HID;

  const int TB = 256;
  const int gNH   = (NH + TB - 1) / TB;
  const int gN    = (N + TB - 1) / TB;
  const int gE    = (E + TB - 1) / TB;
  const int gScat = (int)(((long long)E * 16 + TB - 1) / TB);
  const int gGemm = (N + 31) / 32;
  const int gP1   = ((K1P >> 1) * HID + TB - 1) / TB;
  const int gP2   = ((K2P >> 1) * HID + TB - 1) / TB;
  const bool exact = (N % 32) == 0;     // true for N=100000

  // pack + zero-pad weights (tiny, once per launch)
  pack_weights<<<gP1, TB, 0, stream>>>(W1l, Wp1l, K1, K1P);
  pack_weights<<<gP1, TB, 0, stream>>>(W1r, Wp1r, K1, K1P);
  pack_weights<<<gP2, TB, 0, stream>>>(W2l, Wp2l, K2, K2P);
  pack_weights<<<gP2, TB, 0, stream>>>(W2r, Wp2r, K2, K2P);

  // in-degree (shared by both layers)
  zero_f32<<<gN, TB, 0, stream>>>(CNT, N);
  degree_kernel<<<gE, TB, 0, stream>>>(dst, CNT, E);

  // ---- layer 1: project first (mean commutes with the linear map) ----
  if (exact)
    gemm_dual_wmma<K1, K1P, false><<<gGemm, TB, 0, stream>>>(x, Wp1l, Wp1r, A, B, N);
  else
    gemm_dual_wmma<K1, K1P, true><<<gGemm, TB, 0, stream>>>(x, Wp1l, Wp1r, A, B, N);
  zero_f32<<<gNH, TB, 0, stream>>>(C, NH);
  scatter_feat<<<gScat, TB, 0, stream>>>(A, src, dst, C, E);
  combine_relu<<<gNH, TB, 0, stream>>>(C, CNT, b1, B, H, N);

  // ---- layer 2 ----
  if (exact)
    gemm_dual_wmma<K2, K2P, false><<<gGemm, TB, 0, stream>>>(H, Wp2l, Wp2r, A, B, N);
  else
    gemm_dual_wmma<K2, K2P, true><<<gGemm, TB, 0, stream>>>(H, Wp2l, Wp2r, A, B, N);
  zero_f32<<<gNH, TB, 0, stream>>>(C, NH);
  scatter_feat<<<gScat, TB, 0, stream>>>(A, src, dst, C, E);
  combine_relu<<<gNH, TB, 0, stream>>>(C, CNT, b2, B, H, N);

  // ---- classifier ----
  classifier<<<gN, TB, 0, stream>>>(H, Wc, bc, (float*)d_out, N);
}